// pit_78280073937597
// MI455X (gfx1250) — compile-verified
//
#include <hip/hip_runtime.h>
#include <math.h>

// ---------------------------------------------------------------------------
// PIT pipeline for MI455X (gfx1250, wave32).
// - Never materialize distance/attention matrices: recompute d2 from 2-D
//   coords, fused mask->exp->normalize->matvec (all live data L2-resident).
// - Per-row quantile via in-LDS radix select on float bit patterns.
// - Dense [N,64]x[64,64] GEMMs via V_WMMA_F32_16X16X4_F32 (full f32 fidelity).
// - Attention val tiles staged LDS-double-buffered via the CDNA5 async DMA
//   path (global_load_async_to_lds_b128 + s_wait_asynccnt), overlapped with
//   the distance/exp phase.
// ---------------------------------------------------------------------------

#define NX   11271     // 221*51 input points
#define NL   2886      // 111*26 latent points
#define HID  64

typedef __attribute__((ext_vector_type(2))) float v2f;
typedef __attribute__((ext_vector_type(8))) float v8f;

__device__ __forceinline__ float gelu_erf(float v) {
    // torch F.gelu default: 0.5*x*(1+erf(x/sqrt(2)))
    return 0.5f * v * (1.0f + erff(v * 0.70710678118654752f));
}

// ---------------------------------------------------------------------------
// Build latent coords: ltt = x[:, ::2, ::2, :][:, :111, :26, :]
// ---------------------------------------------------------------------------
__global__ void k_ltt(const float* __restrict__ x, float* __restrict__ ltt) {
    int l = blockIdx.x * blockDim.x + threadIdx.x;
    if (l >= NL) return;
    int i = l / 26, j = l % 26;
    const float* p = x + ((size_t)(2 * i) * 51 + (size_t)(2 * j)) * 2;
    ltt[2 * l + 0] = p[0];
    ltt[2 * l + 1] = p[1];
}

// ---------------------------------------------------------------------------
// Encoder: h0[j,c] = gelu(x0*W[0,c] + x1*W[1,c] + b[c]),  K=2 -> plain VALU
// ---------------------------------------------------------------------------
__global__ void k_encoder(const float* __restrict__ x, const float* __restrict__ W,
                          const float* __restrict__ b, float* __restrict__ h0, int n) {
    int i = blockIdx.x * blockDim.x + threadIdx.x;
    if (i >= n * HID) return;
    int j = i >> 6, c = i & 63;
    float v = fmaf(x[2 * j + 0], W[c], fmaf(x[2 * j + 1], W[HID + c], b[c]));
    h0[i] = gelu_erf(v);
}

// ---------------------------------------------------------------------------
// Repack per-head weights [4][64][16] -> concatenated B [64][64] so the val
// projection becomes a single 64x64 GEMM: B[j][h*16+k] = w[h][j][k]
// ---------------------------------------------------------------------------
__global__ void k_repack(const float* __restrict__ w, float* __restrict__ Bp) {
    int t = blockIdx.x * blockDim.x + threadIdx.x;
    if (t >= HID * HID) return;
    int j = t >> 6, c = t & 63;
    int h = c >> 4, k = c & 15;
    Bp[t] = w[h * (HID * 16) + j * 16 + k];
}

// ---------------------------------------------------------------------------
// Generic WMMA GEMM: C[N,64] = act( A[N,64] @ B[64,64] + bias (+addend) )
// One wave = one 16x16 output tile (wave w of 4 owns cols 16w..16w+15).
// f32 WMMA fragment layouts per CDNA5 ISA 7.12.2:
//   A 16x4 : lanes 0-15 -> M=lane, VGPR{0,1}=K{0,1};  lanes 16-31 -> K{2,3}
//   B 4x16 : lanes 0-15 -> N=lane, VGPR{0,1}=K{0,1};  lanes 16-31 -> K{2,3}
//   C 16x16: VGPR r -> M = r + 8*(lane>=16), N = lane&15
// ---------------------------------------------------------------------------
__global__ void k_gemm_n64(const float* __restrict__ A, const float* __restrict__ B,
                           const float* __restrict__ bias, const float* __restrict__ addend,
                           float* __restrict__ C, int N, int act) {
    int lane = threadIdx.x & 31;
    int wave = threadIdx.x >> 5;           // 0..3 -> column tile
    int r0 = blockIdx.x * 16;
    int colBase = wave * 16;
    int laneN = lane & 15;
    int hi = lane >> 4;                    // half-wave selector
    int kk = hi * 2;                       // K sub-offset {0,2}

    int aRow = r0 + laneN;                 // A-fragment row (M = lane&15)
    if (aRow > N - 1) aRow = N - 1;        // clamp tail reads (stores predicated)
    const float* Arow = A + (size_t)aRow * HID;

    v8f acc = {};
    for (int k = 0; k < HID; k += 4) {
        v2f a, b;
        a.x = Arow[k + kk + 0];
        a.y = Arow[k + kk + 1];
        b.x = B[(k + kk + 0) * HID + colBase + laneN];
        b.y = B[(k + kk + 1) * HID + colBase + laneN];
        acc = __builtin_amdgcn_wmma_f32_16x16x4_f32(
            /*neg_a=*/false, a, /*neg_b=*/false, b,
            /*c_mod=*/(short)0, acc, /*reuse_a=*/false, /*reuse_b=*/false);
    }

    int col = colBase + laneN;
    float bv = bias ? bias[col] : 0.0f;
#pragma unroll
    for (int r = 0; r < 8; r++) {
        int row = r0 + r + 8 * hi;
        if (row < N) {
            float v = acc[r] + bv;
            if (addend) v += addend[(size_t)row * HID + col];
            if (act) v = gelu_erf(v);
            C[(size_t)row * HID + col] = v;
        }
    }
}

// ---------------------------------------------------------------------------
// Per-row quantile of squared distances via in-LDS radix select.
// thr[row] = s[kth] + frac*(s[kth+1] - s[kth])   (jnp.quantile 'linear')
// d2 >= 0 so float bit patterns are order-preserving as uints.
// Dynamic LDS: uint bits[nK] + uint hist[256]
// ---------------------------------------------------------------------------
__global__ void k_quantile(const float* __restrict__ Qc, const float* __restrict__ Kc,
                           int nK, int kth, float frac, float* __restrict__ thr_out) {
    extern __shared__ unsigned int sm[];
    unsigned int* bits = sm;
    unsigned int* hist = sm + nK;
    __shared__ unsigned int s_prefix, s_want, s_cntle, s_mingt;

    int row = blockIdx.x, tid = threadIdx.x;
    float qx = Qc[2 * row + 0], qy = Qc[2 * row + 1];
    for (int j = tid; j < nK; j += blockDim.x) {
        float dx = qx - Kc[2 * j + 0];
        float dy = qy - Kc[2 * j + 1];
        float d2 = fmaf(dx, dx, dy * dy);
        bits[j] = __float_as_uint(d2);
    }
    if (tid == 0) { s_prefix = 0u; s_want = (unsigned)kth; }
    __syncthreads();

    for (int shift = 24; shift >= 0; shift -= 8) {
        for (int b = tid; b < 256; b += blockDim.x) hist[b] = 0u;
        __syncthreads();
        unsigned pfx = s_prefix;
        for (int j = tid; j < nK; j += blockDim.x) {
            unsigned u = bits[j];
            bool match = (shift == 24) || (((u ^ pfx) >> (shift + 8)) == 0u);
            if (match) atomicAdd(&hist[(u >> shift) & 255u], 1u);
        }
        __syncthreads();
        if (tid == 0) {
            unsigned want = s_want, accum = 0u, bsel = 255u;
            for (unsigned b = 0; b < 256u; b++) {
                unsigned c = hist[b];
                if (accum + c > want) { bsel = b; break; }
                accum += c;
            }
            s_prefix = pfx | (bsel << shift);
            s_want = want - accum;
        }
        __syncthreads();
    }

    unsigned v = s_prefix;                 // bit pattern of s[kth]
    if (tid == 0) { s_cntle = 0u; s_mingt = 0xFFFFFFFFu; }
    __syncthreads();
    unsigned cl = 0u, mg = 0xFFFFFFFFu;
    for (int j = tid; j < nK; j += blockDim.x) {
        unsigned u = bits[j];
        if (u <= v) cl++;
        else if (u < mg) mg = u;
    }
    atomicAdd(&s_cntle, cl);
    atomicMin(&s_mingt, mg);
    __syncthreads();
    if (tid == 0) {
        float sk  = __uint_as_float(v);
        float sk1 = (s_cntle >= (unsigned)(kth + 2)) ? sk : __uint_as_float(s_mingt);
        thr_out[row] = fmaf(frac, sk1 - sk, sk);
    }
}

// ---------------------------------------------------------------------------
// Async DMA: stage a 64-row x 64-col f32 val tile (16 KB) into LDS.
// 64 threads x 16 b128 ops x 16 B = 16 KB; ASYNCcnt +16 per wave.
// NOTE: caller guarantees the 16 KB window stays inside the val allocation
// (val is always the NX*64 workspace buffer) and 16-B alignment.
// ---------------------------------------------------------------------------
__device__ __forceinline__ void async_stage_tile(const float* val, int row0,
                                                 void* ldsbuf, int t) {
    unsigned lbase = (unsigned)(size_t)ldsbuf;
    unsigned long long gbase =
        (unsigned long long)(size_t)val + (unsigned long long)row0 * (HID * 4);
#pragma unroll
    for (int i = 0; i < 16; i++) {
        unsigned loff = lbase + (unsigned)((i * 64 + t) * 16);
        unsigned long long ga = gbase + (unsigned long long)((i * 64 + t) * 16);
        asm volatile("global_load_async_to_lds_b128 %0, %1, off"
                     :: "v"(loff), "v"(ga) : "memory");
    }
}

// ---------------------------------------------------------------------------
// Fused masked multi-head softmax attention + gelu.
// out[q, h*16+v] = gelu( sum_j w_h(q,j) * val[j, h*16+v] / sum_j w_h(q,j) )
//   w_h = exp(-c_h * d2(q,j)) if d2 <= thr[q] else 0 ;  c_h = tan(pi/4*(1+sin r_h))
// (positive c_h => head-independent mask; exponents in [-c*thr, 0] => stable.)
// Block = 64 threads (2 waves): thread t owns output column t. Keys processed
// in chunks of 64; the val tile for chunk c+1 is async-DMA'd into the other
// LDS buffer while chunk c's weights are computed (phase 1). Async loads
// complete in order, so s_wait_asynccnt 16 releases the current tile.
// ---------------------------------------------------------------------------
__global__ void k_attn(const float* __restrict__ Qc,
                       const float* __restrict__ Kc, int nK,
                       const float* __restrict__ val,
                       const float* __restrict__ rr,
                       const float* __restrict__ thr,
                       float* __restrict__ out) {
    __shared__ __align__(16) float vtile[2][64][64];   // 32 KB double buffer
    __shared__ float wsh[4][64];
    __shared__ float sden[4][64];
    __shared__ float cs[4];
    __shared__ float sd[4];

    int t = threadIdx.x;                   // 0..63
    int q = blockIdx.x;
    if (t < 4) cs[t] = tanf(0.78539816339744831f * (1.0f + sinf(rr[t])));

    // prologue: stage chunk 0 into vtile[0]
    async_stage_tile(val, 0, &vtile[0][0][0], t);
    __syncthreads();                       // cs visible

    float qx = Qc[2 * q + 0], qy = Qc[2 * q + 1];
    float thrq = thr ? thr[q] : 3.402823466e+38f;
    float c0 = cs[0], c1 = cs[1], c2 = cs[2], c3 = cs[3];
    int myH = t >> 4;
    float num = 0.0f;
    float dp0 = 0.f, dp1 = 0.f, dp2 = 0.f, dp3 = 0.f;

    int p = 0;
    for (int j0 = 0; j0 < nK; j0 += 64, p ^= 1) {
        // both waves done reading vtile[p^1] (phase 2 of previous chunk) and
        // done reading wsh before we overwrite either
        __syncthreads();
        bool hasNext = (j0 + 64) < nK;
        if (hasNext) async_stage_tile(val, j0 + 64, &vtile[p ^ 1][0][0], t);

        // ---- phase 1: distances -> masked per-head weights ----
        int j = j0 + t;
        float w0 = 0.f, w1 = 0.f, w2 = 0.f, w3 = 0.f;
        if (j < nK) {
            float dx = qx - Kc[2 * j + 0];
            float dy = qy - Kc[2 * j + 1];
            float d2 = fmaf(dx, dx, dy * dy);
            if (d2 <= thrq) {
                w0 = expf(-c0 * d2); w1 = expf(-c1 * d2);
                w2 = expf(-c2 * d2); w3 = expf(-c3 * d2);
            }
        }
        wsh[0][t] = w0; wsh[1][t] = w1; wsh[2][t] = w2; wsh[3][t] = w3;
        dp0 += w0; dp1 += w1; dp2 += w2; dp3 += w3;

        // current tile's 16 async ops are the oldest outstanding; in-order
        // completion means cnt<=16 (next issued) / cnt==0 (last chunk) is safe
        if (hasNext) asm volatile("s_wait_asynccnt 0x10" ::: "memory");
        else         asm volatile("s_wait_asynccnt 0x0"  ::: "memory");
        __syncthreads();                   // DMA'd tile + wsh visible to both waves

        // ---- phase 2: accumulate num from LDS tile (bank-conflict-free) ----
        int lim = nK - j0; if (lim > 64) lim = 64;
        for (int jj = 0; jj < lim; jj++)
            num = fmaf(wsh[myH][jj], vtile[p][jj][t], num);
    }

    sden[0][t] = dp0; sden[1][t] = dp1; sden[2][t] = dp2; sden[3][t] = dp3;
    __syncthreads();
    if (t < 4) {
        float s = 0.f;
        for (int i = 0; i < 64; i++) s += sden[t][i];
        sd[t] = s;
    }
    __syncthreads();
    out[(size_t)q * HID + t] = gelu_erf(num / sd[myH]);
}

// ---------------------------------------------------------------------------
// Decoder head: out[j] = t[j,:] . de2_w + de2_b   ([64,1] weight)
// ---------------------------------------------------------------------------
__global__ void k_dot64(const float* __restrict__ t, const float* __restrict__ w,
                        const float* __restrict__ b, float* __restrict__ out, int n) {
    int j = blockIdx.x * blockDim.x + threadIdx.x;
    if (j >= n) return;
    float s = b[0];
    const float* row = t + (size_t)j * HID;
#pragma unroll
    for (int k = 0; k < HID; k++) s = fmaf(row[k], w[k], s);
    out[j] = s;
}

// ---------------------------------------------------------------------------
extern "C" void kernel_launch(void* const* d_in, const int* in_sizes, int n_in,
                              void* d_out, int out_size, void* d_ws, size_t ws_size,
                              hipStream_t stream) {
    (void)in_sizes; (void)n_in; (void)out_size; (void)ws_size;

    const float* x      = (const float*)d_in[0];   // [1,221,51,2] == xf [11271,2]
    const float* en_w   = (const float*)d_in[1];
    const float* en_b   = (const float*)d_in[2];
    const float* down_r = (const float*)d_in[3];
    const float* down_w = (const float*)d_in[4];
    const float* pa_r   = (const float*)d_in[5];   // [4,1,4,1,1]
    const float* pa_w   = (const float*)d_in[6];   // [4,4,64,16]
    const float* mlp1_w = (const float*)d_in[7];
    const float* mlp1_b = (const float*)d_in[8];
    const float* mlp2_w = (const float*)d_in[9];
    const float* mlp2_b = (const float*)d_in[10];
    const float* lin_w  = (const float*)d_in[11];
    const float* lin_b  = (const float*)d_in[12];
    const float* up_r   = (const float*)d_in[13];
    const float* up_w   = (const float*)d_in[14];
    const float* de1_w  = (const float*)d_in[15];
    const float* de1_b  = (const float*)d_in[16];
    const float* de2_w  = (const float*)d_in[17];
    const float* de2_b  = (const float*)d_in[18];

    // workspace layout (floats) ~12.5 MB; vbuf always holds attention val and
    // is sized NX*64 so async tile staging never leaves the allocation.
    float* ws   = (float*)d_ws;
    float* ltt  = ws;                    // NL*2
    float* h0   = ltt  + (size_t)NL * 2; // NX*64
    float* vbuf = h0   + (size_t)NX * HID;
    float* hq   = vbuf + (size_t)NX * HID;
    float* hA   = hq   + (size_t)NX * HID;   // NL*64
    float* hB   = hA   + (size_t)NL * HID;
    float* aL   = hB   + (size_t)NL * HID;
    float* tL   = aL   + (size_t)NL * HID;
    float* mL   = tL   + (size_t)NL * HID;
    float* thrD = mL   + (size_t)NL * HID;   // NL
    float* thrU = thrD + NL;                 // NX
    float* Bp   = thrU + NX;                 // 64*64

    const int gemmNL = (NL + 15) / 16, gemmNX = (NX + 15) / 16;

    // latent coords + encoder
    k_ltt<<<(NL + 255) / 256, 256, 0, stream>>>(x, ltt);
    k_encoder<<<(NX * HID + 255) / 256, 256, 0, stream>>>(x, en_w, en_b, h0, NX);

    // ---- down: latent <- input (5% locality) ----
    k_repack<<<16, 256, 0, stream>>>(down_w, Bp);
    k_gemm_n64<<<gemmNX, 128, 0, stream>>>(h0, Bp, nullptr, nullptr, vbuf, NX, 0);
    // quantile over nK=11271: pos = 0.05*11270 = 563.5 -> k=563, frac=0.5
    k_quantile<<<NL, 256, (size_t)NX * 4 + 1024, stream>>>(ltt, x, NX, 563, 0.5f, thrD);
    k_attn<<<NL, 64, 0, stream>>>(ltt, x, NX, vbuf, down_r, thrD, hA);

    // ---- 4 latent blocks (dense attention, no mask) ----
    float* hin = hA;
    float* hout = hB;
    for (int i = 0; i < 4; i++) {
        k_repack<<<16, 256, 0, stream>>>(pa_w + (size_t)i * 4096, Bp);
        k_gemm_n64<<<gemmNL, 128, 0, stream>>>(hin, Bp, nullptr, nullptr, vbuf, NL, 0);
        k_attn<<<NL, 64, 0, stream>>>(ltt, ltt, NL, vbuf, pa_r + (size_t)i * 4, nullptr, aL);
        k_gemm_n64<<<gemmNL, 128, 0, stream>>>(aL, mlp1_w + (size_t)i * 4096,
                                               mlp1_b + (size_t)i * 64, nullptr, tL, NL, 1);
        k_gemm_n64<<<gemmNL, 128, 0, stream>>>(tL, mlp2_w + (size_t)i * 4096,
                                               mlp2_b + (size_t)i * 64, nullptr, mL, NL, 0);
        // h = gelu(m + h @ lin_w + lin_b)   (ping-pong to avoid in-place hazard)
        k_gemm_n64<<<gemmNL, 128, 0, stream>>>(hin, lin_w + (size_t)i * 4096,
                                               lin_b + (size_t)i * 64, mL, hout, NL, 1);
        float* tmp = hin; hin = hout; hout = tmp;
    }

    // ---- up: input <- latent (5% locality) ----
    k_repack<<<16, 256, 0, stream>>>(up_w, Bp);
    k_gemm_n64<<<gemmNL, 128, 0, stream>>>(hin, Bp, nullptr, nullptr, vbuf, NL, 0);
    // quantile over nK=2886: pos = 0.05*2885 = 144.25 -> k=144, frac=0.25
    k_quantile<<<NX, 256, (size_t)NL * 4 + 1024, stream>>>(x, ltt, NL, 144, 0.25f, thrU);
    k_attn<<<NX, 64, 0, stream>>>(x, ltt, NL, vbuf, up_r, thrU, hq);

    // ---- decoder ----
    k_gemm_n64<<<gemmNX, 128, 0, stream>>>(hq, de1_w, de1_b, nullptr, h0, NX, 1);
    k_dot64<<<(NX + 255) / 256, 256, 0, stream>>>(h0, de2_w, de2_b, (float*)d_out, NX);
}